// LocalConcatSheafLearnerVariant_31842887533247
// MI455X (gfx1250) — compile-verified
//
#include <hip/hip_runtime.h>
#include <cmath>

typedef __attribute__((ext_vector_type(2))) float v2f;
typedef __attribute__((ext_vector_type(8))) float v8f;

#define HIDDEN 32
#define TWOH   64
#define DD     3
#define GPW    16           // groups per wave (one 16x16 WMMA tile)
#define ROW_DW 68           // padded LDS row stride in dwords (bank-conflict free, 8B aligned)
#define WAVES_PER_BLOCK 8

__global__ __launch_bounds__(256)
void sheaf_wmma_kernel(const float* __restrict__ x,
                       const long long* __restrict__ src,
                       const long long* __restrict__ dst,
                       const float* __restrict__ W,
                       float* __restrict__ out,
                       int E, int G)
{
    __shared__ float lds[WAVES_PER_BLOCK][GPW * ROW_DW];

    const int lane = threadIdx.x & 31;
    const int wave = threadIdx.x >> 5;
    const long long tile = (long long)blockIdx.x * WAVES_PER_BLOCK + wave;
    const long long g0   = tile * GPW;            // first group of this tile

    // ---------------- B operand: W^T [64 x 16], sliced for v_wmma_f32_16x16x4_f32.
    // B slice k layout: VGPR0 = (K=4k + 2*(lane/16), N=lane%16), VGPR1 = next K.
    const int N    = lane & 15;
    const int koff = (lane >> 4) * 2;
    float B0[16], B1[16];
#pragma unroll
    for (int k = 0; k < 16; ++k) {
        float bx = 0.f, by = 0.f;
        if (N < 9) {
            const float* wp = W + (long long)N * TWOH + 4 * k + koff;
            bx = wp[0];
            by = wp[1];
        }
        B0[k] = bx; B1[k] = by;
    }

    // ---------------- Load the 48 src / 48 dst indices for this tile (clamped for tail).
    const long long eBase = g0 * DD;
    const long long cap   = (long long)E - 1;
    long long eA = eBase + lane;        if (eA > cap) eA = cap;
    long long eB = eBase + 32 + lane;   if (eB > cap) eB = cap;
    const int srcA = (int)src[eA];
    const int srcB = (int)src[eB];
    const int dstA = (int)dst[eA];
    const int dstB = (int)dst[eB];

    // ---------------- Gather + 3-edge accumulate. Lane owns column `lane` (src half)
    // and column `32+lane` (dst half). Row index is wave-uniform (readlane) so each
    // row gather is a single fully coalesced 128B load.
    float* myLds = lds[wave];
#pragma unroll
    for (int g = 0; g < GPW; ++g) {
        float aS = 0.f, aD = 0.f;
#pragma unroll
        for (int i = 0; i < DD; ++i) {
            const int e = DD * g + i;   // 0..47, compile-time constant after unroll
            const int si = (e < 32) ? __builtin_amdgcn_readlane(srcA, e)
                                    : __builtin_amdgcn_readlane(srcB, e - 32);
            const int di = (e < 32) ? __builtin_amdgcn_readlane(dstA, e)
                                    : __builtin_amdgcn_readlane(dstB, e - 32);
            aS += x[(long long)si * HIDDEN + lane];
            aD += x[(long long)di * HIDDEN + lane];
        }
        myLds[g * ROW_DW + lane]      = aS;
        myLds[g * ROW_DW + 32 + lane] = aD;
    }
    // LDS region is private to this wave and DS ops are in-order per wave:
    // no workgroup barrier needed before reading back.

    // ---------------- WMMA: D[16x16] = A[16x64] * B[64x16], K in 16 slices of 4.
    // A slice k layout: lanes 0-15 hold (M=lane, K=4k..4k+1); lanes 16-31 hold
    // (M=lane-16, K=4k+2..4k+3) -> one 8B LDS read per lane per slice.
    v8f c = {};
    const int aRow = lane & 15;
#pragma unroll
    for (int k = 0; k < 16; ++k) {
        const float* ap = myLds + aRow * ROW_DW + 4 * k + koff;
        v2f a; a.x = ap[0]; a.y = ap[1];
        v2f b; b.x = B0[k]; b.y = B1[k];
#if __has_builtin(__builtin_amdgcn_wmma_f32_16x16x4_f32)
        c = __builtin_amdgcn_wmma_f32_16x16x4_f32(false, a, false, b,
                                                  (short)0, c, false, false);
#else
        // Fallback (should not trigger on gfx1250): scalar FMA emulation of the
        // same accumulation so the kernel still compiles.
        c[0] += a.x * b.x + a.y * b.y;
#endif
    }

    // ---------------- tanh + store. D layout: VGPR r holds (M = r + 8*(lane/16), N = lane%16).
    const int mBase = (lane >> 4) * 8;
#pragma unroll
    for (int r = 0; r < 8; ++r) {
        const long long g = g0 + mBase + r;
        if (N < 9 && g < (long long)G) {
            out[g * 9 + N] = tanhf(c[r]);
        }
    }
}

extern "C" void kernel_launch(void* const* d_in, const int* in_sizes, int n_in,
                              void* d_out, int out_size, void* d_ws, size_t ws_size,
                              hipStream_t stream) {
    const float*     x  = (const float*)d_in[0];
    const long long* ei = (const long long*)d_in[1];   // int64 [2, E]
    const float*     W  = (const float*)d_in[2];       // [9, 64]
    float*           out = (float*)d_out;              // [G, 3, 3] -> G*9 floats

    const int E = in_sizes[1] / 2;
    const int G = out_size / 9;

    const long long* src = ei;
    const long long* dst = ei + E;

    const int tiles  = (G + GPW - 1) / GPW;
    const int blocks = (tiles + WAVES_PER_BLOCK - 1) / WAVES_PER_BLOCK;

    sheaf_wmma_kernel<<<blocks, 256, 0, stream>>>(x, src, dst, W, out, E, G);
}